// RelationDecoder_50302656971024
// MI455X (gfx1250) — compile-verified
//
#include <hip/hip_runtime.h>
#include <hip/hip_bf16.h>
#include <cstdint>

// ---------------------------------------------------------------------------
// RelationDecoder for MI455X (gfx1250, wave32, WMMA).
// fp32 end-to-end on V_WMMA_F32_16X16X4_F32, 16x32 wave tiles (2 accs share
// one A fragment). T tensors streamed HBM->LDS exactly once via double-
// buffered GLOBAL_LOAD_ASYNC_TO_LDS_B128 (ASYNCcnt pipelined against WMMA).
// XOR-swizzled LDS panes -> bank-conflict-free at stride 256, letting
// 2 panes + 2 chunk buffers fill exactly 320 KB/WGP.
// ---------------------------------------------------------------------------

typedef __attribute__((ext_vector_type(2))) float v2f;
typedef __attribute__((ext_vector_type(8))) float v8f;

#define SEQ     512
#define DMODEL  768
#define NCLS    5
#define NENT    128
#define SPAN    16
#define TWOH    256
#define NBC     65536   // 256*256 flattened (b,c)
#define LDT     772     // padded LDS stride for 768-wide rows (gates kernel)

__device__ __forceinline__ v8f wmma_f32(v2f a, v2f b, v8f c) {
  return __builtin_amdgcn_wmma_f32_16x16x4_f32(
      false, a, false, b, (short)0, c, false, false);
}

// Swizzled pane index: stride 256, col ^ 4*(row&15). 16-row column walks
// (A/B fragment reads, lane = row) hit 16 distinct banks; (c,c+1) pairs stay
// adjacent because the XOR only touches bits >= 2.
__device__ __forceinline__ int sw(int row, int col) {
  return (row << 8) + (col ^ ((row & 15) << 2));
}
// Chunk swizzle: [256 rows][32 cols], col bit4 ^= row bit1 -> full-wave
// conflict-free B reads (upper half-wave reads row+2).
__device__ __forceinline__ int cidx(int row, int col) {
  return (row << 5) + (col ^ (((row >> 1) & 1) << 4));
}

// CDNA5 async copy: global -> LDS, 16B per lane, tracked by ASYNCcnt.
__device__ __forceinline__ void async_copy_b128(void* lds_dst, const void* gsrc) {
  uint32_t off = (uint32_t)(uintptr_t)lds_dst;  // low 32 bits = LDS offset
  asm volatile("global_load_async_to_lds_b128 %0, %1, off"
               :: "v"(off), "v"(gsrc) : "memory");
}
#define WAIT_ASYNC(n) asm volatile("s_wait_asynccnt " #n ::: "memory")

// ---------------------------------------------------------------------------
// Stage 1a: LSTM gate GEMM with token gather (unchanged from prev round).
// ---------------------------------------------------------------------------
struct GateParams {
  const float* enc;
  const int*   idx[3];
  const float* Wf[3];
  const float* Wb[3];
  const float* bf[3];
  const float* bb[3];
  float*       gates;   // [3][2048][1024]
};

__global__ void k_gates(GateParams p) {
  extern __shared__ float smem[];
  int*   tks = (int*)smem;        // 64 token indices
  float* tok = smem + 64;         // [64][LDT]
  const int bid = blockIdx.x;
  const int g   = bid >> 9;
  const int rem = bid & 511;
  const int e4  = rem >> 4;       // 4-entity band
  const int nt  = rem & 15;       // 64-wide n tile
  const int tid = threadIdx.x;    // 256 threads

  if (tid < 64) tks[tid] = p.idx[g][e4 * 64 + tid];
  __syncthreads();
  for (int i = tid; i < 64 * DMODEL; i += 256) {
    int s = i / DMODEL, d = i - s * DMODEL;
    tok[s * LDT + d] = p.enc[(size_t)tks[s] * DMODEL + d];
  }
  __syncthreads();

  const int wave = tid >> 5, lane = tid & 31, ln = lane & 15;
  const int h2 = (lane >> 4) << 1, hf8 = (lane >> 4) << 3;
  const int m0 = (wave & 3) * 16;
  const int n0 = nt * 64 + (wave >> 2) * 32;
  const bool fwd = (n0 < 512);
  const float* W = fwd ? p.Wf[g] : p.Wb[g];
  const float* B = fwd ? p.bf[g] : p.bb[g];
  const int nc   = n0 - (fwd ? 0 : 512);

  v8f acc0 = {}, acc1 = {};
  for (int k0 = 0; k0 < DMODEL; k0 += 4) {
    v2f a;  a.x  = tok[(m0 + ln) * LDT + k0 + h2];
            a.y  = tok[(m0 + ln) * LDT + k0 + h2 + 1];
    v2f b0; b0.x = W[(size_t)(nc + ln) * DMODEL + k0 + h2];
            b0.y = W[(size_t)(nc + ln) * DMODEL + k0 + h2 + 1];
    v2f b1; b1.x = W[(size_t)(nc + 16 + ln) * DMODEL + k0 + h2];
            b1.y = W[(size_t)(nc + 16 + ln) * DMODEL + k0 + h2 + 1];
    acc0 = wmma_f32(a, b0, acc0);
    acc1 = wmma_f32(a, b1, acc1);
  }
  const float bias0 = B[nc + ln];
  const float bias1 = B[nc + 16 + ln];
  float* out = p.gates + ((size_t)g * 2048 + (size_t)e4 * 64) * 1024;
  for (int r = 0; r < 8; ++r) {
    int row = m0 + r + hf8;
    out[(size_t)row * 1024 + n0 + ln]      = acc0[r] + bias0;
    out[(size_t)row * 1024 + n0 + 16 + ln] = acc1[r] + bias1;
  }
}

// ---------------------------------------------------------------------------
// Stage 1b: LSTM nonlinearity + max-pool over span.
// ---------------------------------------------------------------------------
__global__ void k_pool(const float* __restrict__ gates, float* __restrict__ Eall) {
  const int g   = blockIdx.x >> 7;
  const int ent = blockIdx.x & 127;
  const int t   = threadIdx.x;
  const int dir = t >> 7, hd = t & 127;
  const float* base = gates + ((size_t)g * 2048 + (size_t)ent * SPAN) * 1024 + dir * 512;
  float m = -3.4e38f;
  for (int s = 0; s < SPAN; ++s) {
    const float* row = base + (size_t)s * 1024;
    float gi = row[hd];
    float gg = row[256 + hd];
    float go = row[384 + hd];
    float c = (1.f / (1.f + expf(-gi))) * tanhf(gg);
    float h = (1.f / (1.f + expf(-go))) * tanhf(c);
    m = fmaxf(m, h);
  }
  Eall[(size_t)g * (NENT * TWOH) + (size_t)ent * TWOH + t] = m;
}

// ---------------------------------------------------------------------------
// Fused bilinear stage, one block per middle index b (grid=256):
//   GEMM1: Qb[i,c] = sum_a EL[i,a] * T[a, b*256+c]   -> X pane (LDS)
//   GEMM2: out[b][e][i] = sum_c ER[e,c] * Qb[i,c]
// T slice streamed via double-buffered async-to-LDS 32-col chunks.
// LDS: 2 swizzled panes (2x128KB) + 2 chunks (2x32KB) = 320KB exactly.
// ---------------------------------------------------------------------------
__global__ void k_bilinear(const float* __restrict__ EL, const float* __restrict__ T,
                           const float* __restrict__ ER, float* __restrict__ out) {
  extern __shared__ float smem[];
  float* S1  = smem;                 // pane [128][256] swizzled (EL, then ER)
  float* X   = smem + 32768;         // pane [128][256] swizzled (Qb)
  float* Bl0 = smem + 65536;         // chunk [256][32]
  float* Bl1 = smem + 65536 + 8192;  // chunk [256][32]
  const int tid = threadIdx.x;
  const int b   = blockIdx.x;
  const float* Tb = T + (size_t)b * 256;   // element (a,c) = Tb[a*NBC + c]

  for (int t = tid; t < NENT * TWOH; t += 256) {
    int r = t >> 8, c = t & 255;
    S1[sw(r, c)] = EL[t];
  }
  // Kick off chunk 0 while the pane settles (8 async b128 per wave).
  for (int t = tid; t < 2048; t += 256) {
    int aa = t >> 3, cc4 = (t & 7) << 2;
    async_copy_b128(Bl0 + cidx(aa, cc4), Tb + (size_t)aa * NBC + cc4);
  }
  __syncthreads();

  const int wave = tid >> 5, lane = tid & 31, ln = lane & 15;
  const int h2 = (lane >> 4) << 1, hf8 = (lane >> 4) << 3;
  const int m0 = wave * 16;

  // ---- GEMM1: 8 double-buffered chunks of 32 columns ---------------------
  for (int cb = 0; cb < TWOH; cb += 32) {
    float* cur = ((cb >> 5) & 1) ? Bl1 : Bl0;
    float* nxt = ((cb >> 5) & 1) ? Bl0 : Bl1;
    if (cb + 32 < TWOH) {
      for (int t = tid; t < 2048; t += 256) {
        int aa = t >> 3, cc4 = (t & 7) << 2;
        async_copy_b128(nxt + cidx(aa, cc4), Tb + (size_t)aa * NBC + cb + 32 + cc4);
      }
      if (cb + 64 < TWOH)   // warm L2 two chunks ahead
        __builtin_prefetch(&Tb[(size_t)tid * NBC + cb + 64], 0, 1);
      WAIT_ASYNC(8);        // chunk cb complete (in-order), cb+32 in flight
    } else {
      WAIT_ASYNC(0);
    }
    __syncthreads();

    v8f acc0 = {}, acc1 = {};
    for (int k0 = 0; k0 < TWOH; k0 += 4) {
      const int kb = k0 + h2;
      const int xr = ((kb >> 1) & 1) << 4;
      const int c0l = ln ^ xr, c1l = (16 + ln) ^ xr;
      int ab = sw(m0 + ln, kb);
      v2f a;  a.x  = S1[ab];              a.y  = S1[ab + 1];
      v2f b0; b0.x = cur[(kb << 5) + c0l]; b0.y = cur[((kb + 1) << 5) + c0l];
      v2f b1; b1.x = cur[(kb << 5) + c1l]; b1.y = cur[((kb + 1) << 5) + c1l];
      acc0 = wmma_f32(a, b0, acc0);
      acc1 = wmma_f32(a, b1, acc1);
    }
    for (int r = 0; r < 8; ++r) {
      X[sw(m0 + r + hf8, cb + ln)]      = acc0[r];
      X[sw(m0 + r + hf8, cb + 16 + ln)] = acc1[r];
    }
    __syncthreads();   // release cur for reuse two iterations later
  }

  // ---- swap A pane to ER --------------------------------------------------
  for (int t = tid; t < NENT * TWOH; t += 256) {
    int r = t >> 8, c = t & 255;
    S1[sw(r, c)] = ER[t];
  }
  __syncthreads();

  // ---- GEMM2: out[e,i] = sum_c ER[e,c]*Qb[i,c] ----------------------------
  float* outb = out + (size_t)b * (NENT * NENT);
  for (int nt = 0; nt < NENT; nt += 32) {
    v8f acc0 = {}, acc1 = {};
    for (int c0 = 0; c0 < TWOH; c0 += 4) {
      int ab = sw(m0 + ln, c0 + h2);
      int x0 = sw(nt + ln, c0 + h2);
      int x1 = sw(nt + 16 + ln, c0 + h2);
      v2f a;  a.x  = S1[ab]; a.y  = S1[ab + 1];
      v2f b0; b0.x = X[x0];  b0.y = X[x0 + 1];
      v2f b1; b1.x = X[x1];  b1.y = X[x1 + 1];
      acc0 = wmma_f32(a, b0, acc0);
      acc1 = wmma_f32(a, b1, acc1);
    }
    for (int r = 0; r < 8; ++r) {
      int e = m0 + r + hf8;
      outb[(size_t)e * NENT + nt + ln]      = acc0[r];
      outb[(size_t)e * NENT + nt + 16 + ln] = acc1[r];
    }
  }
}

// ---------------------------------------------------------------------------
// Stage 3, one block per (k,m) (grid=640):
//   X[j,c] = sum_b tep_k[j,b] * T_cls[b,m,c]
//   pred[i,j,k,m] = sum_c hep_k[i,c] * X[j,c]
// Same async double-buffered streaming for the T_cls slice. pred scatter
// (stride 5 floats) merges in the 192MB L2 (pred fits entirely).
// ---------------------------------------------------------------------------
__global__ void k_stage3(const float* __restrict__ hep, const float* __restrict__ tep,
                         const float* __restrict__ Tcls, float* __restrict__ pred) {
  extern __shared__ float smem[];
  float* S1  = smem;                 // pane: tep_k, then hep_k (swizzled)
  float* X   = smem + 32768;         // pane
  float* Bl0 = smem + 65536;
  float* Bl1 = smem + 65536 + 8192;
  const int tid = threadIdx.x;
  const int kk  = blockIdx.x / NCLS;
  const int m   = blockIdx.x % NCLS;

  for (int t = tid; t < NENT * TWOH; t += 256) {
    int j = t & 127, bb = t >> 7;
    S1[sw(j, bb)] = tep[((size_t)bb * NENT + kk) * NENT + j];
  }
  for (int t = tid; t < 2048; t += 256) {
    int aa = t >> 3, cc4 = (t & 7) << 2;
    async_copy_b128(Bl0 + cidx(aa, cc4),
                    Tcls + ((size_t)aa * NCLS + m) * TWOH + cc4);
  }
  __syncthreads();

  const int wave = tid >> 5, lane = tid & 31, ln = lane & 15;
  const int h2 = (lane >> 4) << 1, hf8 = (lane >> 4) << 3;
  const int m0 = wave * 16;

  // ---- Phase 1: X = tep_k @ T_cls[:,m,:] ----------------------------------
  for (int cb = 0; cb < TWOH; cb += 32) {
    float* cur = ((cb >> 5) & 1) ? Bl1 : Bl0;
    float* nxt = ((cb >> 5) & 1) ? Bl0 : Bl1;
    if (cb + 32 < TWOH) {
      for (int t = tid; t < 2048; t += 256) {
        int aa = t >> 3, cc4 = (t & 7) << 2;
        async_copy_b128(nxt + cidx(aa, cc4),
                        Tcls + ((size_t)aa * NCLS + m) * TWOH + cb + 32 + cc4);
      }
      if (cb + 64 < TWOH)
        __builtin_prefetch(&Tcls[((size_t)tid * NCLS + m) * TWOH + cb + 64], 0, 1);
      WAIT_ASYNC(8);
    } else {
      WAIT_ASYNC(0);
    }
    __syncthreads();

    v8f acc0 = {}, acc1 = {};
    for (int b0i = 0; b0i < TWOH; b0i += 4) {
      const int kb = b0i + h2;
      const int xr = ((kb >> 1) & 1) << 4;
      const int c0l = ln ^ xr, c1l = (16 + ln) ^ xr;
      int ab = sw(m0 + ln, kb);
      v2f a;  a.x  = S1[ab];              a.y  = S1[ab + 1];
      v2f b0; b0.x = cur[(kb << 5) + c0l]; b0.y = cur[((kb + 1) << 5) + c0l];
      v2f b1; b1.x = cur[(kb << 5) + c1l]; b1.y = cur[((kb + 1) << 5) + c1l];
      acc0 = wmma_f32(a, b0, acc0);
      acc1 = wmma_f32(a, b1, acc1);
    }
    for (int r = 0; r < 8; ++r) {
      X[sw(m0 + r + hf8, cb + ln)]      = acc0[r];
      X[sw(m0 + r + hf8, cb + 16 + ln)] = acc1[r];
    }
    __syncthreads();
  }

  // ---- swap S1 <- hep_k ---------------------------------------------------
  for (int t = tid; t < NENT * TWOH; t += 256) {
    int i = t & 127, cc = t >> 7;
    S1[sw(i, cc)] = hep[((size_t)cc * NENT + kk) * NENT + i];
  }
  __syncthreads();

  // ---- Phase 2: pred[i,j,k,m] = sum_c hep_k[i,c] * X[j,c] -----------------
  for (int nt = 0; nt < NENT; nt += 32) {
    v8f acc0 = {}, acc1 = {};
    for (int c0 = 0; c0 < TWOH; c0 += 4) {
      int ab = sw(m0 + ln, c0 + h2);
      int x0 = sw(nt + ln, c0 + h2);
      int x1 = sw(nt + 16 + ln, c0 + h2);
      v2f a;  a.x  = S1[ab]; a.y  = S1[ab + 1];
      v2f b0; b0.x = X[x0];  b0.y = X[x0 + 1];
      v2f b1; b1.x = X[x1];  b1.y = X[x1 + 1];
      acc0 = wmma_f32(a, b0, acc0);
      acc1 = wmma_f32(a, b1, acc1);
    }
    for (int r = 0; r < 8; ++r) {
      int i = m0 + r + hf8;
      pred[(((size_t)i * NENT + nt + ln)      * NENT + kk) * NCLS + m] = acc0[r];
      pred[(((size_t)i * NENT + nt + 16 + ln) * NENT + kk) * NCLS + m] = acc1[r];
    }
  }
}

// ---------------------------------------------------------------------------
extern "C" void kernel_launch(void* const* d_in, const int* in_sizes, int n_in,
                              void* d_out, int out_size, void* d_ws, size_t ws_size,
                              hipStream_t stream) {
  (void)in_sizes; (void)n_in; (void)out_size; (void)ws_size;
  const float* enc  = (const float*)d_in[0];
  const float* The  = (const float*)d_in[16];
  const float* Tte  = (const float*)d_in[17];
  const float* Tcl  = (const float*)d_in[18];

  // Workspace layout (floats). Total ~59 MB.
  float* ws    = (float*)d_ws;
  float* Eall  = ws;                                    // 3*128*256
  float* gates = Eall + 3 * NENT * TWOH;                // 3*2048*1024
  float* hep   = gates + (size_t)3 * 2048 * 1024;       // [256][128][128]
  float* tep   = hep + (size_t)TWOH * NENT * NENT;      // [256][128][128]

  GateParams p;
  p.enc = enc;
  p.idx[0] = (const int*)d_in[1];
  p.idx[1] = (const int*)d_in[2];
  p.idx[2] = (const int*)d_in[3];
  p.Wf[0] = (const float*)d_in[4];  p.bf[0] = (const float*)d_in[5];
  p.Wb[0] = (const float*)d_in[6];  p.bb[0] = (const float*)d_in[7];
  p.Wf[1] = (const float*)d_in[8];  p.bf[1] = (const float*)d_in[9];
  p.Wb[1] = (const float*)d_in[10]; p.bb[1] = (const float*)d_in[11];
  p.Wf[2] = (const float*)d_in[12]; p.bf[2] = (const float*)d_in[13];
  p.Wb[2] = (const float*)d_in[14]; p.bb[2] = (const float*)d_in[15];
  p.gates = gates;

  const size_t lds_g = (64 + 64 * LDT) * sizeof(float);   // ~193 KB
  const size_t lds_b = (size_t)81920 * sizeof(float);     // exactly 320 KB

  (void)hipFuncSetAttribute((const void*)k_gates,
      hipFuncAttributeMaxDynamicSharedMemorySize, (int)lds_g);
  (void)hipFuncSetAttribute((const void*)k_bilinear,
      hipFuncAttributeMaxDynamicSharedMemorySize, (int)lds_b);
  (void)hipFuncSetAttribute((const void*)k_stage3,
      hipFuncAttributeMaxDynamicSharedMemorySize, (int)lds_b);

  // Stream-ordered pipeline.
  k_gates   <<<dim3(1536), dim3(256), lds_g, stream>>>(p);
  k_pool    <<<dim3(384),  dim3(256), 0,     stream>>>(gates, Eall);
  k_bilinear<<<dim3(256),  dim3(256), lds_b, stream>>>(Eall + 0 * NENT * TWOH, The,
                                                       Eall + 2 * NENT * TWOH, hep);
  k_bilinear<<<dim3(256),  dim3(256), lds_b, stream>>>(Eall + 1 * NENT * TWOH, Tte,
                                                       Eall + 2 * NENT * TWOH, tep);
  k_stage3  <<<dim3(640),  dim3(256), lds_b, stream>>>(hep, tep, Tcl, (float*)d_out);
}